// SMultiHeadAttention_17489106829911
// MI455X (gfx1250) — compile-verified
//
#include <hip/hip_runtime.h>

typedef float v2f __attribute__((ext_vector_type(2)));
typedef float v8f __attribute__((ext_vector_type(8)));
typedef unsigned int u32x4 __attribute__((ext_vector_type(4)));
typedef int i32x4 __attribute__((ext_vector_type(4)));
typedef int i32x8 __attribute__((ext_vector_type(8)));

constexpr int Bb = 2, Ss = 12, Nn = 256, Ee = 512, Hh = 8, DKk = 64, SHh = 96;
constexpr int NS = Nn * Ss;                            // 3072
constexpr size_t QKV_SZ = (size_t)Bb * Ss * Nn * Ee;   // 3,145,728 (== ctx output size)
constexpr size_t EB_SZ  = (size_t)Bb * Nn * Nn * SHh;  // 12,582,912

#if __has_builtin(__builtin_amdgcn_tensor_load_to_lds) && __has_builtin(__builtin_amdgcn_s_wait_tensorcnt)
#define USE_TDM 1
#else
#define USE_TDM 0
#endif

static __device__ __forceinline__ v8f wmma4(v2f a, v2f b, v8f c) {
  // V_WMMA_F32_16X16X4_F32 : D = A(16x4,f32) x B(4x16,f32) + C(16x16,f32)
  return __builtin_amdgcn_wmma_f32_16x16x4_f32(false, a, false, b, (short)0, c, false, false);
}
static __device__ __forceinline__ v8f vzero8() {
  v8f z = {0.f,0.f,0.f,0.f,0.f,0.f,0.f,0.f};
  return z;
}

constexpr int ilog2c(int v) { int l = 0; while (v > 1) { v >>= 1; ++l; } return l; }

// ---------------------------------------------------------------------------
// Load a ROWS x COLS f32 tile from global (row stride gstride dwords) into LDS
// (row stride LSTRIDE dwords).  On gfx1250 with the TDM builtin available this
// is a single tensor_load_to_lds DMA issued by wave 0 (TENSORcnt tracked);
// LSTRIDE == COLS+1 padding is expressed with D# pad_enable/pad_interval.
// Otherwise: cooperative copy by all 256 threads.
// ---------------------------------------------------------------------------
template <int ROWS, int COLS, int LSTRIDE>
static __device__ __forceinline__ void tile_load(float* lds, const float* g, int gstride) {
#if USE_TDM
  constexpr bool PADDED = (LSTRIDE == COLS + 1);
  constexpr bool TDM_OK =
      (LSTRIDE == COLS) ||
      (PADDED && ((COLS & (COLS - 1)) == 0) && COLS >= 2 && COLS <= 256);
  if constexpr (TDM_OK) {
    if (threadIdx.x < 32) {   // one wave issues the DMA (EXEC ignored by TDM)
      unsigned lds_off = (unsigned)(unsigned long long)lds;  // LDS aperture: low 32 bits = byte offset
      unsigned long long ga = (unsigned long long)g;
      // D# group 0: [1:0]=count(1), [63:32]=lds_addr, [120:64]=global_addr, [127:126]=type(2)
      u32x4 g0;
      g0[0] = 1u;
      g0[1] = lds_off;
      g0[2] = (unsigned)(ga & 0xffffffffull);
      g0[3] = (unsigned)((ga >> 32) & 0x01ffffffull) | (2u << 30);
      // D# group 1: data_size=2(4B); optional pad 1 dword every COLS dwords;
      // tensor_dim0 = gstride (>= tile width), tensor_dim1 huge; tile dims; dim0 stride.
      unsigned w0 = (2u << 16);
      if constexpr (PADDED)
        w0 |= (1u << 20) | ((unsigned)(ilog2c(COLS) - 1) << 22);  // pad_amount=0 -> 1 dword
      const unsigned td0 = (unsigned)gstride;
      const unsigned td1 = 0x40000000u;
      i32x8 g1;
      g1[0] = (int)w0;
      g1[1] = (int)((td0 & 0xffffu) << 16);
      g1[2] = (int)(((td0 >> 16) & 0xffffu) | ((td1 & 0xffffu) << 16));
      g1[3] = (int)(((td1 >> 16) & 0xffffu) | ((unsigned)COLS << 16));
      g1[4] = ROWS;                       // tile_dim1 (tile_dim2 = 0)
      g1[5] = (int)(unsigned)gstride;     // tensor_dim0_stride[31:0]
      g1[6] = 0;
      g1[7] = 0;
      i32x4 zz = {0, 0, 0, 0};
      i32x8 z8 = {0, 0, 0, 0, 0, 0, 0, 0};
      __builtin_amdgcn_tensor_load_to_lds(g0, g1, zz, zz, z8, 0);
    }
    return;
  }
#endif
  for (int idx = threadIdx.x; idx < ROWS * COLS; idx += 256) {
    int r = idx / COLS, c = idx % COLS;
    lds[r * LSTRIDE + c] = g[(size_t)r * gstride + c];
  }
}

static __device__ __forceinline__ void tile_fence() {
#if USE_TDM
  if (threadIdx.x < 32) __builtin_amdgcn_s_wait_tensorcnt((short)0);
#endif
  __syncthreads();
}

// ---------------------------------------------------------------------------
// K1: Q/K/V projection.  in is [B,E,N,S] -> per b an E x (N*S) matrix (K-major).
// out[b,s,n,:] = (in[b,:,n,s])^T @ W.   M=N*S=3072, K=E=512, Nout=E=512.
// 128x64 workgroup tile, 2x2 register blocking per wave (4 WMMA / 8 lds dwords).
// ---------------------------------------------------------------------------
__global__ __launch_bounds__(256) void k_qkv_proj(const float* __restrict__ in,
                                                  const float* __restrict__ W,
                                                  float* __restrict__ out) {
  __shared__ float As[64][128];  // [k][m] (A^T tile; frag reads run along m -> conflict-free)
  __shared__ float Bs[64][64];   // [k][e]
  const int m0 = blockIdx.x * 128;
  const int e0 = blockIdx.y * 64;
  const int b  = blockIdx.z;
  const float* Ain = in + (size_t)b * Ee * NS;
  const int lane = threadIdx.x & 31, wave = threadIdx.x >> 5;
  const int hi = lane >> 4, l15 = lane & 15;
  const int wr = wave >> 1, wc = wave & 1;
  const int rm0 = wr * 32 + l15, rm1 = rm0 + 16;
  const int c0 = wc * 32 + l15, c1 = c0 + 16;
  v8f c00 = vzero8(), c01 = vzero8(), c10 = vzero8(), c11 = vzero8();
  for (int k0 = 0; k0 < Ee; k0 += 64) {
    tile_load<64, 128, 128>(&As[0][0], Ain + (size_t)k0 * NS + m0, NS);
    tile_load<64, 64, 64>(&Bs[0][0], W + (size_t)k0 * Ee + e0, Ee);
    tile_fence();
#pragma unroll
    for (int t = 0; t < 16; ++t) {
      int kb = t * 4 + hi * 2;
      v2f a0; a0.x = As[kb][rm0]; a0.y = As[kb + 1][rm0];
      v2f a1; a1.x = As[kb][rm1]; a1.y = As[kb + 1][rm1];
      v2f b0; b0.x = Bs[kb][c0];  b0.y = Bs[kb + 1][c0];
      v2f b1; b1.x = Bs[kb][c1];  b1.y = Bs[kb + 1][c1];
      c00 = wmma4(a0, b0, c00);
      c01 = wmma4(a0, b1, c01);
      c10 = wmma4(a1, b0, c10);
      c11 = wmma4(a1, b1, c11);
    }
    __syncthreads();
  }
  const int rb = wr * 32 + hi * 8;
#pragma unroll
  for (int r = 0; r < 8; ++r) {
    int mA = m0 + rb + r;
    int mB = mA + 16;
    int nA = mA / Ss, sA = mA % Ss;   // m = n*S + s (S contiguous in input)
    int nB = mB / Ss, sB = mB % Ss;
    size_t roA = (((size_t)b * Ss + sA) * Nn + nA) * Ee + e0;
    size_t roB = (((size_t)b * Ss + sB) * Nn + nB) * Ee + e0;
    out[roA + c0] = c00[r];
    out[roA + c1] = c01[r];
    out[roB + c0] = c10[r];
    out[roB + c1] = c11[r];
  }
}

// ---------------------------------------------------------------------------
// K2: EB = input_E @ We.   M = B*N*N = 131072, K = 512, Nout = 96.
// ---------------------------------------------------------------------------
__global__ __launch_bounds__(256) void k_eb_proj(const float* __restrict__ inE,
                                                 const float* __restrict__ We,
                                                 float* __restrict__ EB) {
  __shared__ float As[64][65];   // [m][k]
  __shared__ float Bs[64][96];   // [k][j]
  const int m0 = blockIdx.x * 64;
  const int lane = threadIdx.x & 31, wave = threadIdx.x >> 5;
  const int hi = lane >> 4, l15 = lane & 15;
  const int tr = wave >> 1, tc0 = (wave & 1) * 3;
  v8f c[3] = {vzero8(), vzero8(), vzero8()};
  for (int k0 = 0; k0 < Ee; k0 += 64) {
    tile_load<64, 64, 65>(&As[0][0], inE + (size_t)m0 * Ee + k0, Ee);
    tile_load<64, 96, 96>(&Bs[0][0], We + (size_t)k0 * SHh, SHh);
    tile_fence();
    const int row = tr * 16 + l15;
#pragma unroll
    for (int t = 0; t < 16; ++t) {
      int kb = t * 4 + hi * 2;
      v2f a; a.x = As[row][kb]; a.y = As[row][kb + 1];
#pragma unroll
      for (int u = 0; u < 3; ++u) {
        int col = (tc0 + u) * 16 + l15;
        v2f bb; bb.x = Bs[kb][col]; bb.y = Bs[kb + 1][col];
        c[u] = wmma4(a, bb, c[u]);
      }
    }
    __syncthreads();
  }
  const int rbase = tr * 16 + hi * 8;
  for (int r = 0; r < 8; ++r) {
    size_t ro = (size_t)(m0 + rbase + r) * SHh;
    for (int u = 0; u < 3; ++u) EB[ro + (tc0 + u) * 16 + l15] = c[u][r];
  }
}

// ---------------------------------------------------------------------------
// K3: attention core per (b,s,h), 32-row block of the 256x256 score matrix.
// scores = (Q Kt)*scale ; e = scores * EB[b,:,:,h*S+s] ; scatter e -> EOUT ;
// softmax rows ; ctx = attn @ V.
// ---------------------------------------------------------------------------
__global__ __launch_bounds__(256) void k_attn(const float* __restrict__ Q,
                                              const float* __restrict__ Kt,
                                              const float* __restrict__ V,
                                              const float* __restrict__ EB,
                                              float* __restrict__ EOUT,
                                              float* __restrict__ CTX) {
  __shared__ float Qs[32][65];
  __shared__ float Ks[64][65];     // reused for K then V tiles
  __shared__ float Es[32][257];    // e_/attn rows (full 256 columns), padded
  const int n0 = blockIdx.x * 32;
  int id = blockIdx.y;
  const int h = id % Hh; id /= Hh;
  const int s = id % Ss; id /= Ss;
  const int b = id;
  const int lane = threadIdx.x & 31, wave = threadIdx.x >> 5;
  const int hi = lane >> 4, l15 = lane & 15;
  const int tr = wave >> 2, tc = wave & 3;      // 2x4 tile grid over 32x64
  const size_t base = (((size_t)b * Ss + s) * Nn) * Ee + (size_t)h * DKk;
  tile_load<32, 64, 65>(&Qs[0][0], Q + base + (size_t)n0 * Ee, Ee);
  const int rbase = tr * 16 + hi * 8;
  const int row = tr * 16 + l15;
  const int col = tc * 16 + l15;
  // ---- scores / e_ ----
  for (int mb = 0; mb < 4; ++mb) {
    const int m0 = mb * 64;
    __syncthreads();  // prior Ks readers done before DMA overwrites
    tile_load<64, 64, 65>(&Ks[0][0], Kt + base + (size_t)m0 * Ee, Ee);
    tile_fence();     // Qs (first iter) + Ks visible to all waves
    v8f c = vzero8();
#pragma unroll
    for (int t = 0; t < 16; ++t) {
      int kb = t * 4 + hi * 2;
      v2f a;  a.x  = Qs[row][kb]; a.y  = Qs[row][kb + 1];
      v2f bb; bb.x = Ks[col][kb]; bb.y = Ks[col][kb + 1];   // B = K^T
      c = wmma4(a, bb, c);
    }
#pragma unroll
    for (int r = 0; r < 8; ++r) {
      int ng = n0 + rbase + r;
      int mg = m0 + tc * 16 + l15;
      float val = c[r] * 0.125f *
                  EB[(((size_t)b * Nn + ng) * Nn + mg) * SHh + h * Ss + s];
      EOUT[(((size_t)b * Nn + mg) * Nn + ng) * SHh + s * Hh + h] = val;
      Es[rbase + r][mg] = val;
    }
  }
  __syncthreads();
  // ---- softmax: 8 threads per row, shfl reductions within lane groups of 8 ----
  {
    const int srow = threadIdx.x >> 3;
    const int cb = (threadIdx.x & 7) * 32;
    float mx = -3.0e38f;
    for (int i = 0; i < 32; ++i) mx = fmaxf(mx, Es[srow][cb + i]);
    for (int o = 1; o < 8; o <<= 1) mx = fmaxf(mx, __shfl_xor(mx, o, 32));
    float sum = 0.f;
    for (int i = 0; i < 32; ++i) {
      float v = __expf(Es[srow][cb + i] - mx);
      Es[srow][cb + i] = v;
      sum += v;
    }
    for (int o = 1; o < 8; o <<= 1) sum += __shfl_xor(sum, o, 32);
    float inv = 1.0f / sum;
    for (int i = 0; i < 32; ++i) Es[srow][cb + i] *= inv;
  }
  __syncthreads();
  // ---- ctx = attn @ V ----
  v8f cc = vzero8();
  for (int mb = 0; mb < 4; ++mb) {
    tile_load<64, 64, 65>(&Ks[0][0], V + base + (size_t)(mb * 64) * Ee, Ee);
    tile_fence();
#pragma unroll
    for (int t = 0; t < 16; ++t) {
      int jl = t * 4 + hi * 2;
      int kg = mb * 64 + jl;
      v2f a;  a.x  = Es[row][kg]; a.y  = Es[row][kg + 1];
      v2f bb; bb.x = Ks[jl][col]; bb.y = Ks[jl + 1][col];
      cc = wmma4(a, bb, cc);
    }
    __syncthreads();
  }
  for (int r = 0; r < 8; ++r) {
    CTX[(((size_t)b * Ss + s) * Nn + n0 + rbase + r) * Ee + h * DKk + tc * 16 + l15] = cc[r];
  }
}

// ---------------------------------------------------------------------------
// K4: TMP = CTX @ Wo + bo + xq   (M = B*S*N = 6144, K = Nout = 512)
// 128x64 workgroup tile, 2x2 register blocking.
// ---------------------------------------------------------------------------
__global__ __launch_bounds__(256) void k_wo(const float* __restrict__ CTX,
                                            const float* __restrict__ Wo,
                                            const float* __restrict__ bo,
                                            const float* __restrict__ inQ,
                                            float* __restrict__ TMP) {
  __shared__ float As[128][65];
  __shared__ float Bs[64][64];
  const int m0 = blockIdx.x * 128;
  const int e0 = blockIdx.y * 64;
  const int lane = threadIdx.x & 31, wave = threadIdx.x >> 5;
  const int hi = lane >> 4, l15 = lane & 15;
  const int wr = wave >> 1, wc = wave & 1;
  const int rm0 = wr * 32 + l15, rm1 = rm0 + 16;
  const int c0 = wc * 32 + l15, c1 = c0 + 16;
  v8f c00 = vzero8(), c01 = vzero8(), c10 = vzero8(), c11 = vzero8();
  for (int k0 = 0; k0 < Ee; k0 += 64) {
    tile_load<128, 64, 65>(&As[0][0], CTX + (size_t)m0 * Ee + k0, Ee);
    tile_load<64, 64, 64>(&Bs[0][0], Wo + (size_t)k0 * Ee + e0, Ee);
    tile_fence();
#pragma unroll
    for (int t = 0; t < 16; ++t) {
      int kb = t * 4 + hi * 2;
      v2f a0; a0.x = As[rm0][kb]; a0.y = As[rm0][kb + 1];
      v2f a1; a1.x = As[rm1][kb]; a1.y = As[rm1][kb + 1];
      v2f b0; b0.x = Bs[kb][c0];  b0.y = Bs[kb + 1][c0];
      v2f b1; b1.x = Bs[kb][c1];  b1.y = Bs[kb + 1][c1];
      c00 = wmma4(a0, b0, c00);
      c01 = wmma4(a0, b1, c01);
      c10 = wmma4(a1, b0, c10);
      c11 = wmma4(a1, b1, c11);
    }
    __syncthreads();
  }
  const int rb = wr * 32 + hi * 8;
  for (int r = 0; r < 8; ++r) {
    for (int half = 0; half < 2; ++half) {
      int m = m0 + rb + half * 16 + r;
      int b = m / (Ss * Nn); int rem = m % (Ss * Nn);
      int s = rem / Nn, n = rem % Nn;
      int e1 = e0 + c0, e2 = e0 + c1;
      float v1 = (half ? c10[r] : c00[r]) + bo[e1] + inQ[(((size_t)b * Ee + e1) * Nn + n) * Ss + s];
      float v2 = (half ? c11[r] : c01[r]) + bo[e2] + inQ[(((size_t)b * Ee + e2) * Nn + n) * Ss + s];
      TMP[(size_t)m * Ee + e1] = v1;
      TMP[(size_t)m * Ee + e2] = v2;
    }
  }
}

// ---------------------------------------------------------------------------
// K5: layernorm rows of TMP (512) and transposed store to out [B,E,N,S]
// ---------------------------------------------------------------------------
__global__ __launch_bounds__(256) void k_ln_ctx(const float* __restrict__ TMP,
                                                const float* __restrict__ g,
                                                const float* __restrict__ be,
                                                float* __restrict__ out) {
  const int lane = threadIdx.x & 31, wave = threadIdx.x >> 5;
  const int m = blockIdx.x * 8 + wave;          // < 6144
  const float* x = TMP + (size_t)m * Ee;
  float v[16];
  float sum = 0.f;
  for (int i = 0; i < 16; ++i) { v[i] = x[lane + 32 * i]; sum += v[i]; }
  for (int o = 1; o < 32; o <<= 1) sum += __shfl_xor(sum, o, 32);
  float mu = sum * (1.0f / Ee);
  float vs = 0.f;
  for (int i = 0; i < 16; ++i) { float d = v[i] - mu; vs += d * d; }
  for (int o = 1; o < 32; o <<= 1) vs += __shfl_xor(vs, o, 32);
  float rstd = rsqrtf(vs * (1.0f / Ee) + 1e-5f);
  int b = m / (Ss * Nn); int rem = m % (Ss * Nn);
  int s = rem / Nn, n = rem % Nn;
  for (int i = 0; i < 16; ++i) {
    int e = lane + 32 * i;
    out[(((size_t)b * Ee + e) * Nn + n) * Ss + s] = (v[i] - mu) * rstd * g[e] + be[e];
  }
}

// ---------------------------------------------------------------------------
// K6: outE = EOUT @ Wfe + bfe + input_E   (M = 131072, K = 96, Nout = 512)
// ---------------------------------------------------------------------------
__global__ __launch_bounds__(256) void k_fe(const float* __restrict__ EOUT,
                                            const float* __restrict__ Wfe,
                                            const float* __restrict__ bfe,
                                            const float* __restrict__ inE,
                                            float* __restrict__ outE) {
  __shared__ float As[64][97];   // K=96 per row; stride 97 (manual copy path)
  __shared__ float Bs[96][64];
  const int m0 = blockIdx.x * 64;
  const int e0 = blockIdx.y * 64;
  const int lane = threadIdx.x & 31, wave = threadIdx.x >> 5;
  const int hi = lane >> 4, l15 = lane & 15;
  const int tr = wave >> 1, tc0 = (wave & 1) * 2;
  tile_load<64, 96, 97>(&As[0][0], EOUT + (size_t)m0 * SHh, SHh);  // falls back to copy loop
  tile_load<96, 64, 64>(&Bs[0][0], Wfe + e0, Ee);
  tile_fence();
  v8f c0 = vzero8(), c1 = vzero8();
  const int row = tr * 16 + l15;
#pragma unroll
  for (int t = 0; t < 24; ++t) {
    int kb = t * 4 + hi * 2;
    v2f a; a.x = As[row][kb]; a.y = As[row][kb + 1];
    int ca = tc0 * 16 + l15, cb = (tc0 + 1) * 16 + l15;
    v2f b0; b0.x = Bs[kb][ca]; b0.y = Bs[kb + 1][ca];
    c0 = wmma4(a, b0, c0);
    v2f b1; b1.x = Bs[kb][cb]; b1.y = Bs[kb + 1][cb];
    c1 = wmma4(a, b1, c1);
  }
  const int rbase = tr * 16 + hi * 8;
  for (int r = 0; r < 8; ++r) {
    size_t m = (size_t)(m0 + rbase + r);
    int e1 = e0 + tc0 * 16 + l15, e2 = e1 + 16;
    outE[m * Ee + e1] = c0[r] + bfe[e1] + inE[m * Ee + e1];
    outE[m * Ee + e2] = c1[r] + bfe[e2] + inE[m * Ee + e2];
  }
}

// ---------------------------------------------------------------------------
// K7: in-place layernorm of outE rows (512 each)
// ---------------------------------------------------------------------------
__global__ __launch_bounds__(256) void k_ln_e(float* __restrict__ outE,
                                              const float* __restrict__ g,
                                              const float* __restrict__ be) {
  const int lane = threadIdx.x & 31, wave = threadIdx.x >> 5;
  const size_t m = (size_t)blockIdx.x * 8 + wave;   // < 131072
  float* x = outE + m * Ee;
  float v[16];
  float sum = 0.f;
  for (int i = 0; i < 16; ++i) { v[i] = x[lane + 32 * i]; sum += v[i]; }
  for (int o = 1; o < 32; o <<= 1) sum += __shfl_xor(sum, o, 32);
  float mu = sum * (1.0f / Ee);
  float vs = 0.f;
  for (int i = 0; i < 16; ++i) { float d = v[i] - mu; vs += d * d; }
  for (int o = 1; o < 32; o <<= 1) vs += __shfl_xor(vs, o, 32);
  float rstd = rsqrtf(vs * (1.0f / Ee) + 1e-5f);
  for (int i = 0; i < 16; ++i) {
    int e = lane + 32 * i;
    x[e] = (v[i] - mu) * rstd * g[e] + be[e];
  }
}

extern "C" void kernel_launch(void* const* d_in, const int* in_sizes, int n_in,
                              void* d_out, int out_size, void* d_ws, size_t ws_size,
                              hipStream_t stream) {
  (void)in_sizes; (void)n_in; (void)out_size; (void)ws_size;
  const float* inQ = (const float*)d_in[0];
  const float* inK = (const float*)d_in[1];
  const float* inV = (const float*)d_in[2];
  const float* inE = (const float*)d_in[3];
  const float* Wq  = (const float*)d_in[4];
  const float* Wk  = (const float*)d_in[5];
  const float* Wv  = (const float*)d_in[6];
  const float* We  = (const float*)d_in[7];
  const float* Wo  = (const float*)d_in[8];
  const float* bo  = (const float*)d_in[9];
  const float* Wfe = (const float*)d_in[10];
  const float* bfe = (const float*)d_in[11];
  const float* g_o = (const float*)d_in[12];
  const float* b_o = (const float*)d_in[13];
  const float* g_e = (const float*)d_in[14];
  const float* b_e = (const float*)d_in[15];

  float* ws   = (float*)d_ws;
  float* Q    = ws;
  float* K    = ws + QKV_SZ;
  float* V    = ws + 2 * QKV_SZ;
  float* CTX  = ws + 3 * QKV_SZ;
  float* EB   = ws + 4 * QKV_SZ;
  float* EOUT = EB + EB_SZ;

  float* out  = (float*)d_out;
  float* outE = out + QKV_SZ;   // ctx region is B*E*N*S == QKV_SZ floats

  dim3 blk(256);
  // 1) Q/K/V projections (M=3072, Nout=512, per batch), 128x64 tiles
  k_qkv_proj<<<dim3(NS / 128, Ee / 64, Bb), blk, 0, stream>>>(inQ, Wq, Q);
  k_qkv_proj<<<dim3(NS / 128, Ee / 64, Bb), blk, 0, stream>>>(inK, Wk, K);
  k_qkv_proj<<<dim3(NS / 128, Ee / 64, Bb), blk, 0, stream>>>(inV, Wv, V);
  // 2) EB = input_E @ We
  k_eb_proj<<<dim3((Bb * Nn * Nn) / 64), blk, 0, stream>>>(inE, We, EB);
  // 3) attention core
  k_attn<<<dim3(Nn / 32, Bb * Ss * Hh), blk, 0, stream>>>(Q, K, V, EB, EOUT, CTX);
  // 4) output projection + residual (TMP reuses Q buffer)
  k_wo<<<dim3((Bb * Ss * Nn) / 128, Ee / 64), blk, 0, stream>>>(CTX, Wo, bo, inQ, Q);
  // 5) layernorm + transpose to [B,E,N,S]
  k_ln_ctx<<<dim3((Bb * Ss * Nn) / 8), blk, 0, stream>>>(Q, g_o, b_o, out);
  // 6) e-path FFN + residual
  k_fe<<<dim3((Bb * Nn * Nn) / 64, Ee / 64), blk, 0, stream>>>(EOUT, Wfe, bfe, inE, outE);
  // 7) in-place layernorm of e output
  k_ln_e<<<dim3((Bb * Nn * Nn) / 8), blk, 0, stream>>>(outE, g_e, b_e);
}